// SSMConv_79044578115782
// MI455X (gfx1250) — compile-verified
//
#include <hip/hip_runtime.h>
#include <hip/hip_bf16.h>

typedef __attribute__((ext_vector_type(2))) float v2f;
typedef __attribute__((ext_vector_type(8))) float v8f;

#define SEQ    4096
#define DM     1024
#define NSTATE 64
#define NC     16      // channels per workgroup
#define NT     256     // output timesteps per workgroup
#define WIN    320     // staged input window per channel
#define UPITCH 340     // 320 + 320/16 pad floats (swizzle idx + (idx>>4))
#define KPITCH 68      // 64 + pad
#define OPITCH 257     // 256 + pad

#if defined(__gfx1250__) && __has_builtin(__builtin_amdgcn_global_load_async_to_lds_b32)
#define ASYNC_STAGE 1
#else
#define ASYNC_STAGE 0
#endif

typedef __attribute__((address_space(1))) int gbl_i32;   // global-AS int (per hipcc diagnostic)
typedef __attribute__((address_space(3))) int lds_i32;   // LDS-AS int

__device__ __forceinline__ void wait_async_zero() {
#if ASYNC_STAGE
#if __has_builtin(__builtin_amdgcn_s_wait_asynccnt)
    __builtin_amdgcn_s_wait_asynccnt(0);
#else
    asm volatile("s_wait_asynccnt 0" ::: "memory");
#endif
#endif
}

// ---------- prologue 1: csum[s] = sum_m C[m,s] ----------
__global__ void csum_kernel(const float* __restrict__ C, float* __restrict__ csum) {
    int s = threadIdx.x;               // 64 threads
    float acc = 0.f;
    for (int m = 0; m < DM; ++m) acc += C[m * NSTATE + s];
    csum[s] = acc;
}

// ---------- prologue 2: Kt[l*DM + c] = sum_s csum[s]*B[s,c]*exp(A[s]*dt[c]*l) ----------
__global__ void kt_kernel(const float* __restrict__ A, const float* __restrict__ Bp,
                          const float* __restrict__ log_dt, const float* __restrict__ csum,
                          float* __restrict__ Kt) {
    int idx = blockIdx.x * blockDim.x + threadIdx.x;   // 64*1024 threads
    int c = idx & (DM - 1);
    int l = idx >> 10;
    float dt = __expf(log_dt[c]);
    float acc = 0.f;
    for (int s = 0; s < NSTATE; ++s) {
        acc += csum[s] * Bp[s * DM + c] * __expf(A[s] * dt * (float)l);
    }
    Kt[l * DM + c] = acc;
}

// ---------- main: depthwise 64-tap conv as Toeplitz-band WMMA f32 ----------
__launch_bounds__(256)
__global__ void ssm_conv_wmma(const float* __restrict__ u,
                              const float* __restrict__ Kt,    // [64][1024]
                              const float* __restrict__ Dvec,  // [1024]
                              float* __restrict__ y) {
    __shared__ float u_lds[NC * UPITCH];
    __shared__ float k_lds[NC * KPITCH];
    __shared__ float o_lds[NC * OPITCH];

    const int tid  = threadIdx.x;
    const int lane = tid & 31;
    const int wave = tid >> 5;           // 8 waves
    const int cl   = tid & 15;           // staging: channel lane
    const int tq   = tid >> 4;           // staging: time slot 0..15

    const int T0 = blockIdx.x * NT;      // output time base
    const int c0 = blockIdx.y * NC;      // channel base
    const int b  = blockIdx.z;           // batch
    const int W0 = T0 - 32;              // input window base

    // ---- stage u window [W0, W0+320), coalesced across channels ----
    const float* ubase = u + (size_t)b * SEQ * DM + c0 + cl;
    auto stage_u = [&](int tl, bool check) {
        int t    = W0 + tl;
        int lidx = cl * UPITCH + tl + (tl >> 4);       // pad swizzle
        bool ok  = !check || ((unsigned)t < (unsigned)SEQ);
#if ASYNC_STAGE
        if (ok) {
            __builtin_amdgcn_global_load_async_to_lds_b32(
                (gbl_i32*)(ubase + (size_t)t * DM),
                (lds_i32*)&u_lds[lidx], 0, 0);
        } else {
            u_lds[lidx] = 0.f;
        }
#else
        float v = 0.f;
        if (ok) v = ubase[(size_t)t * DM];
        u_lds[lidx] = v;
#endif
    };
    if (W0 >= 0 && W0 + WIN <= SEQ) {
        #pragma unroll
        for (int p = 0; p < WIN / 16; ++p) stage_u(p * 16 + tq, false);
    } else {
        #pragma unroll
        for (int p = 0; p < WIN / 16; ++p) stage_u(p * 16 + tq, true);
    }
    // ---- stage per-channel kernels ----
    #pragma unroll
    for (int p = 0; p < 4; ++p) {
        int l = p * 16 + tq;
#if ASYNC_STAGE
        __builtin_amdgcn_global_load_async_to_lds_b32(
            (gbl_i32*)&Kt[l * DM + c0 + cl],
            (lds_i32*)&k_lds[cl * KPITCH + l], 0, 0);
#else
        k_lds[cl * KPITCH + l] = Kt[l * DM + c0 + cl];
#endif
    }
    wait_async_zero();
    __syncthreads();

    const int row = lane & 15;           // A row a' / B column m / D column
    const int hi  = lane >> 4;           // half-wave selects K-pair {0,1} vs {2,3}

    for (int cc = wave * 2; cc < wave * 2 + 2; ++cc) {   // 2 channels per wave
        const float* uc = &u_lds[cc * UPITCH];
        const float* kp = &k_lds[cc * KPITCH];
        v8f acc = {0.f, 0.f, 0.f, 0.f, 0.f, 0.f, 0.f, 0.f};

        // one MMA K-chunk: A[a',j]=u[T0+16a'-32+j], Bmat[j,m]=K[m+63-j]
        auto mma_step = [&](int kc, bool edge_lo, bool edge_hi) {
            int j0   = 4 * kc + 2 * hi;
            int aidx = 16 * row + j0;                    // even, (aidx&15)<=14
            int ap0  = aidx + (aidx >> 4);
            v2f afrag = { uc[ap0], uc[ap0 + 1] };
            int l0 = row + 63 - j0;                      // K tap for vgpr0 row
            int l1 = l0 - 1;                             // K tap for vgpr1 row
            float b0 = kp[l0 & 63];
            float b1 = kp[l1 & 63];
            if (edge_hi) {                               // kc<4: taps may exceed 63
                if (l0 > 63) b0 = 0.f;
                if (l1 > 63) b1 = 0.f;
            }
            if (edge_lo) {                               // kc>=16: taps may go negative
                if (l0 < 0) b0 = 0.f;
                if (l1 < 0) b1 = 0.f;
            }
            v2f bfrag = { b0, b1 };
            acc = __builtin_amdgcn_wmma_f32_16x16x4_f32(
                      false, afrag, false, bfrag, (short)0, acc, false, false);
        };
        #pragma unroll
        for (int kc = 0; kc < 4; ++kc)  mma_step(kc, false, true);   // upper band edge
        #pragma unroll
        for (int kc = 4; kc < 16; ++kc) mma_step(kc, false, false);  // interior: no checks
        #pragma unroll
        for (int kc = 16; kc < 20; ++kc) mma_step(kc, true, false);  // lower band edge

        // epilogue: + D[c]*u, transpose into LDS for coalesced store
        float Dc = Dvec[c0 + cc];
        float* oc = &o_lds[cc * OPITCH];
        #pragma unroll
        for (int r = 0; r < 8; ++r) {
            int ap   = r + 8 * hi;                 // D-matrix row (output block a')
            int tl   = 16 * ap + row;              // local output time
            int uidx = tl + 32;                    // u[T0+tl] inside staged window
            float uval = uc[uidx + (uidx >> 4)];
            oc[tl] = acc[r] + Dc * uval;
        }
    }
    __syncthreads();

    // ---- coalesced global store: lanes across channels ----
    #pragma unroll
    for (int p = 0; p < NT / 16; ++p) {
        int tl = p * 16 + tq;
        y[((size_t)b * SEQ + T0 + tl) * DM + c0 + cl] = o_lds[cl * OPITCH + tl];
    }
}

extern "C" void kernel_launch(void* const* d_in, const int* in_sizes, int n_in,
                              void* d_out, int out_size, void* d_ws, size_t ws_size,
                              hipStream_t stream) {
    const float* u      = (const float*)d_in[0];
    const float* A      = (const float*)d_in[1];
    const float* Bp     = (const float*)d_in[2];
    const float* Cp     = (const float*)d_in[3];
    const float* Dp     = (const float*)d_in[4];
    const float* log_dt = (const float*)d_in[5];
    float* y = (float*)d_out;

    float* csum = (float*)d_ws;                       // 64 floats
    float* Kt   = (float*)((char*)d_ws + 1024);       // 64*1024 floats

    csum_kernel<<<1, 64, 0, stream>>>(Cp, csum);
    kt_kernel<<<(NSTATE * DM) / 256, 256, 0, stream>>>(A, Bp, log_dt, csum, Kt);

    dim3 grid(SEQ / NT, DM / NC, 8);
    ssm_conv_wmma<<<grid, 256, 0, stream>>>(u, Kt, Dp, y);
}